// EnhancedGNN_61753039782323
// MI455X (gfx1250) — compile-verified
//
#include <hip/hip_runtime.h>
#include <hip/hip_bf16.h>
#include <math.h>

// Problem constants (match reference)
#define NN 20000
#define EE 320000
#define GG 32
#define LL 4
#define TT 4
#define HH 64
#define INF_ 128
#define FF 16
#define HJ 320           // H*(L+1)
#define KPOST 208        // 13*F

typedef float v2f __attribute__((ext_vector_type(2)));
typedef float v8f __attribute__((ext_vector_type(8)));

// ---- fp32 WMMA 16x16x4 wrapper (D = A*B + C), wave32 ------------------------
__device__ __forceinline__ v8f wmma4(v2f a, v2f b, v8f c) {
  // 8 args: (neg_a, A, neg_b, B, c_mod, C, reuse_a, reuse_b)
  return __builtin_amdgcn_wmma_f32_16x16x4_f32(false, a, false, b, (short)0, c,
                                               false, false);
}

// ---- order-preserving float<->uint keys for atomic max/min ------------------
__device__ __forceinline__ unsigned fkey(float f) {
  unsigned u = __float_as_uint(f);
  return ((int)u < 0) ? ~u : (u | 0x80000000u);
}
__device__ __forceinline__ float funkey(unsigned k) {
  unsigned u = (k & 0x80000000u) ? (k & 0x7fffffffu) : ~k;
  return __uint_as_float(u);
}

// ============================================================================
// degree / amp
// ============================================================================
__global__ void k_deg(const int* __restrict__ dsts, float* __restrict__ deg) {
  int e = blockIdx.x * blockDim.x + threadIdx.x;
  if (e < EE) atomicAdd(&deg[dsts[e]], 1.0f);
}
__global__ void k_logsum(const float* __restrict__ deg, float* __restrict__ sl) {
  int n = blockIdx.x * blockDim.x + threadIdx.x;
  if (n < NN) atomicAdd(sl, __logf(deg[n] + 1.0f));
}
__global__ void k_amp(const float* __restrict__ deg, const float* __restrict__ sl,
                      float* __restrict__ amp) {
  int n = blockIdx.x * blockDim.x + threadIdx.x;
  if (n < NN) {
    float avg = sl[0] / (float)NN;
    amp[n] = __logf(fmaxf(deg[n], 1.0f) + 1.0f) / avg;
  }
}

// ============================================================================
// encoder: xs[:,0:64] = x(20000x128) @ enc_W(128x64) + enc_b   (WMMA)
// block = 128 threads = 4 waves; wave -> 16-col tile; blockIdx.x -> 16-row tile
// ============================================================================
__global__ void k_encoder(const float* __restrict__ x, const float* __restrict__ W,
                          const float* __restrict__ bias, float* __restrict__ xsall) {
  const int lane = threadIdx.x & 31;
  const int col0 = (threadIdx.x >> 5) * 16;
  const int row0 = blockIdx.x * 16;
  const int m = lane & 15;
  const int kh = lane >> 4;
  const float* arow = x + (size_t)(row0 + m) * INF_;
  v8f c = {};
#pragma unroll
  for (int kk = 0; kk < INF_; kk += 4) {
    int k0 = kk + 2 * kh;
    v2f a, b;
    a.x = arow[k0];
    a.y = arow[k0 + 1];
    b.x = W[(size_t)k0 * HH + col0 + m];
    b.y = W[(size_t)(k0 + 1) * HH + col0 + m];
    c = wmma4(a, b, c);
  }
  float bv = bias[col0 + m];
#pragma unroll
  for (int r = 0; r < 8; ++r)
    xsall[(size_t)(row0 + r + 8 * kh) * HJ + col0 + m] = c[r] + bv;
}

// ============================================================================
// edge pre-MLP fused with segment scatter (sum, sum^2, max, min).
// Per tower t (blockIdx.y): h[16e x 16f] = [x_dst | x_src](16x32) @ pW_t(32x16)
// then atomics into per-node stats. Never materializes the ExHx tensor.
// ============================================================================
__global__ void k_edge(const float* __restrict__ xsall, int xcol0,
                       const int* __restrict__ srcs, const int* __restrict__ dsts,
                       const float* __restrict__ pW, const float* __restrict__ pb,
                       float* __restrict__ s1, float* __restrict__ s2,
                       unsigned* __restrict__ mxk, unsigned* __restrict__ mnk) {
  const int lane = threadIdx.x & 31;
  const int t = blockIdx.y;
  const int e0 = (blockIdx.x * 4 + (threadIdx.x >> 5)) * 16;
  const int m = lane & 15;
  const int kh = lane >> 4;
  const int e = e0 + m;
  // prefetch upcoming edge indices (speculative; OOB silently dropped)
  __builtin_prefetch(&dsts[e + 4096], 0, 1);
  __builtin_prefetch(&srcs[e + 4096], 0, 1);
  const float* drow = xsall + (size_t)dsts[e] * HJ + xcol0 + t * FF;
  const float* srow = xsall + (size_t)srcs[e] * HJ + xcol0 + t * FF;
  const float* Wt = pW + (size_t)t * 32 * FF;
  v8f c = {};
#pragma unroll
  for (int kk = 0; kk < 32; kk += 4) {
    int k0 = kk + 2 * kh;
    v2f a, b;
    a.x = (k0 < 16) ? drow[k0] : srow[k0 - 16];
    a.y = (k0 + 1 < 16) ? drow[k0 + 1] : srow[k0 + 1 - 16];
    b.x = Wt[k0 * FF + m];
    b.y = Wt[(k0 + 1) * FF + m];
    c = wmma4(a, b, c);
  }
  float bv = pb[t * FF + m];
#pragma unroll
  for (int r = 0; r < 8; ++r) {
    int er = e0 + r + 8 * kh;
    float h = c[r] + bv;
    int node = dsts[er];
    int idx = (node * TT + t) * FF + m;
    atomicAdd(&s1[idx], h);
    atomicAdd(&s2[idx], h * h);
    atomicMax(&mxk[idx], fkey(h));
    atomicMin(&mnk[idx], fkey(h));
  }
}

// ============================================================================
// finalize aggregates: agg[n,t,{mean,mn,mx,std}x16]
// ============================================================================
__global__ void k_agg(const float* __restrict__ deg, const float* __restrict__ s1,
                      const float* __restrict__ s2, const unsigned* __restrict__ mxk,
                      const unsigned* __restrict__ mnk, float* __restrict__ aggb) {
  int idx = blockIdx.x * blockDim.x + threadIdx.x;  // over N*64
  if (idx >= NN * HH) return;
  int n = idx >> 6;
  int tf = idx & 63;
  float d = deg[n];
  float dc = fmaxf(d, 1.0f);
  float mean = s1[idx] / dc;
  float var = fmaxf(s2[idx] / dc - mean * mean, 0.0f);
  float stdv = sqrtf(var + 1e-5f);
  bool empty = (d == 0.0f);
  float mx = empty ? 0.0f : funkey(mxk[idx]);
  float mn = empty ? 0.0f : funkey(mnk[idx]);
  int t = tf >> 4, f = tf & 15;
  float* o = aggb + (size_t)(n * TT + t) * 64 + f;
  o[0] = mean; o[16] = mn; o[32] = mx; o[48] = stdv;
}

// ============================================================================
// node post-MLP: A(n,208)=[xt | agg | agg*a | agg/a] per tower; @ qW(208x16)+qb
// ============================================================================
__device__ __forceinline__ float postA(const float* __restrict__ xsall, int xcol0,
                                       const float* __restrict__ aggb,
                                       const float* __restrict__ amp,
                                       int n, int t, int k) {
  if (k < 16) return xsall[(size_t)n * HJ + xcol0 + t * FF + k];
  int k2 = k - 16;
  int grp = k2 >> 6;
  float v = aggb[(size_t)(n * TT + t) * 64 + (k2 & 63)];
  if (grp == 0) return v;
  float a = amp[n];
  return (grp == 1) ? v * a : v / a;
}

__global__ void k_post(const float* __restrict__ xsall, int xcol0,
                       const float* __restrict__ aggb, const float* __restrict__ amp,
                       const float* __restrict__ qW, const float* __restrict__ qb,
                       float* __restrict__ post) {
  const int lane = threadIdx.x & 31;
  const int t = threadIdx.x >> 5;  // tower
  const int row0 = blockIdx.x * 16;
  const int m = lane & 15;
  const int kh = lane >> 4;
  const int n = row0 + m;
  const float* Wt = qW + (size_t)t * KPOST * FF;
  v8f c = {};
#pragma unroll 4
  for (int kk = 0; kk < KPOST; kk += 4) {
    int k0 = kk + 2 * kh;
    v2f a, b;
    a.x = postA(xsall, xcol0, aggb, amp, n, t, k0);
    a.y = postA(xsall, xcol0, aggb, amp, n, t, k0 + 1);
    b.x = Wt[k0 * FF + m];
    b.y = Wt[(k0 + 1) * FF + m];
    c = wmma4(a, b, c);
  }
  float bv = qb[t * FF + m];
#pragma unroll
  for (int r = 0; r < 8; ++r)
    post[(size_t)(row0 + r + 8 * kh) * HH + t * FF + m] = c[r] + bv;
}

// ============================================================================
// linear + BN + ReLU -> next layer columns of xsall
// ============================================================================
__global__ void k_lin(const float* __restrict__ post, const float* __restrict__ lW,
                      const float* __restrict__ lb, const float* __restrict__ gamma,
                      const float* __restrict__ beta, float* __restrict__ xsall,
                      int outcol0) {
  const int lane = threadIdx.x & 31;
  const int col0 = (threadIdx.x >> 5) * 16;
  const int row0 = blockIdx.x * 16;
  const int m = lane & 15;
  const int kh = lane >> 4;
  const float* arow = post + (size_t)(row0 + m) * HH;
  v8f c = {};
#pragma unroll
  for (int kk = 0; kk < HH; kk += 4) {
    int k0 = kk + 2 * kh;
    v2f a, b;
    a.x = arow[k0];
    a.y = arow[k0 + 1];
    b.x = lW[k0 * HH + col0 + m];
    b.y = lW[(k0 + 1) * HH + col0 + m];
    c = wmma4(a, b, c);
  }
  const float bnscale = rsqrtf(1.0f + 1e-5f);
  int cc = col0 + m;
  float g = gamma[cc] * bnscale, bb = beta[cc], lbv = lb[cc];
#pragma unroll
  for (int r = 0; r < 8; ++r) {
    float v = (c[r] + lbv) * g + bb;
    xsall[(size_t)(row0 + r + 8 * kh) * HJ + outcol0 + cc] = fmaxf(v, 0.0f);
  }
}

// ============================================================================
// graph readout
// ============================================================================
__global__ void k_cnt(const int* __restrict__ batch, float* __restrict__ cnt) {
  int n = blockIdx.x * blockDim.x + threadIdx.x;
  if (n < NN) atomicAdd(&cnt[batch[n]], 1.0f);
}
__global__ void k_pool(const float* __restrict__ xsall, const int* __restrict__ batch,
                       float* __restrict__ gsum, unsigned* __restrict__ gmaxk) {
  int idx = blockIdx.x * blockDim.x + threadIdx.x;  // N*320
  if (idx >= NN * HJ) return;
  int n = idx / HJ, c = idx % HJ;
  float v = xsall[idx];
  int g = batch[n];
  atomicAdd(&gsum[g * HJ + c], v);
  atomicMax(&gmaxk[g * HJ + c], fkey(v));
}
__global__ void k_gbuild(const float* __restrict__ gsum, const unsigned* __restrict__ gmaxk,
                         const float* __restrict__ cnt, float* __restrict__ gbuf) {
  int idx = blockIdx.x * blockDim.x + threadIdx.x;  // G*960
  if (idx >= GG * 3 * HJ) return;
  int g = idx / (3 * HJ);
  int cc = idx % (3 * HJ);
  int grp = cc / HJ, c = cc % HJ;
  float cv = cnt[g];
  float v;
  if (grp == 0)       v = gsum[g * HJ + c] / fmaxf(cv, 1.0f);
  else if (grp == 1)  v = gsum[g * HJ + c];
  else                v = (cv == 0.0f) ? 0.0f : funkey(gmaxk[g * HJ + c]);
  gbuf[idx] = v;
}
__global__ void k_mlp1(const float* __restrict__ gbuf, const float* __restrict__ W1,
                       const float* __restrict__ b1, float* __restrict__ hidden) {
  int idx = blockIdx.x * blockDim.x + threadIdx.x;  // G*64
  if (idx >= GG * HH) return;
  int g = idx / HH, o = idx % HH;
  float acc = b1[o];
  const float* gr = gbuf + (size_t)g * 3 * HJ;
  for (int c = 0; c < 3 * HJ; ++c) acc = fmaf(gr[c], W1[(size_t)c * HH + o], acc);
  hidden[idx] = fmaxf(acc, 0.0f);
}
__global__ void k_mlp2(const float* __restrict__ hidden, const float* __restrict__ W2,
                       const float* __restrict__ b2, float* __restrict__ out) {
  int g = threadIdx.x;
  if (g >= GG) return;
  float acc = b2[0];
  const float* hr = hidden + g * HH;
#pragma unroll
  for (int o = 0; o < HH; ++o) acc = fmaf(hr[o], W2[o], acc);
  out[g] = acc;
}

// ============================================================================
extern "C" void kernel_launch(void* const* d_in, const int* in_sizes, int n_in,
                              void* d_out, int out_size, void* d_ws, size_t ws_size,
                              hipStream_t stream) {
  const float* x     = (const float*)d_in[0];
  const int*   ei    = (const int*)d_in[1];
  const int*   batch = (const int*)d_in[2];
  const float* encW  = (const float*)d_in[3];
  const float* encb  = (const float*)d_in[4];
  const float* preW  = (const float*)d_in[5];
  const float* preb  = (const float*)d_in[6];
  const float* postW = (const float*)d_in[7];
  const float* postb = (const float*)d_in[8];
  const float* linW  = (const float*)d_in[9];
  const float* linb  = (const float*)d_in[10];
  const float* gamma = (const float*)d_in[11];
  const float* beta  = (const float*)d_in[12];
  const float* W1    = (const float*)d_in[13];
  const float* b1    = (const float*)d_in[14];
  const float* W2    = (const float*)d_in[15];
  const float* b2    = (const float*)d_in[16];
  float* out = (float*)d_out;

  const int* srcs = ei;
  const int* dsts = ei + EE;

  // ---- workspace carve-out (floats) ----
  float* wsf = (float*)d_ws;
  size_t o = 0;
  float*    deg    = wsf + o; o += NN;
  float*    amp    = wsf + o; o += NN;
  float*    sumlog = wsf + o; o += 32;
  float*    cnt    = wsf + o; o += GG;
  float*    s1     = wsf + o; o += (size_t)NN * HH;
  float*    s2     = wsf + o; o += (size_t)NN * HH;
  unsigned* mxk    = (unsigned*)(wsf + o); o += (size_t)NN * HH;
  unsigned* mnk    = (unsigned*)(wsf + o); o += (size_t)NN * HH;
  float*    aggb   = wsf + o; o += (size_t)NN * TT * 64;
  float*    post   = wsf + o; o += (size_t)NN * HH;
  float*    xsall  = wsf + o; o += (size_t)NN * HJ;
  float*    gsum   = wsf + o; o += (size_t)GG * HJ;
  unsigned* gmaxk  = (unsigned*)(wsf + o); o += (size_t)GG * HJ;
  float*    gbuf   = wsf + o; o += (size_t)GG * 3 * HJ;
  float*    hidden = wsf + o; o += (size_t)GG * HH;

  // ---- reset accumulators every call (graph-capture safe) ----
  hipMemsetAsync(deg, 0, NN * sizeof(float), stream);
  hipMemsetAsync(sumlog, 0, sizeof(float), stream);
  hipMemsetAsync(cnt, 0, GG * sizeof(float), stream);
  hipMemsetAsync(gsum, 0, (size_t)GG * HJ * sizeof(float), stream);
  hipMemsetAsync(gmaxk, 0, (size_t)GG * HJ * sizeof(unsigned), stream);

  k_deg<<<(EE + 255) / 256, 256, 0, stream>>>(dsts, deg);
  k_logsum<<<(NN + 255) / 256, 256, 0, stream>>>(deg, sumlog);
  k_amp<<<(NN + 255) / 256, 256, 0, stream>>>(deg, sumlog, amp);

  k_encoder<<<NN / 16, 128, 0, stream>>>(x, encW, encb, xsall);

  for (int l = 0; l < LL; ++l) {
    hipMemsetAsync(s1, 0, (size_t)NN * HH * sizeof(float), stream);
    hipMemsetAsync(s2, 0, (size_t)NN * HH * sizeof(float), stream);
    hipMemsetAsync(mxk, 0x00, (size_t)NN * HH * sizeof(unsigned), stream);
    hipMemsetAsync(mnk, 0xFF, (size_t)NN * HH * sizeof(unsigned), stream);

    k_edge<<<dim3(EE / 64, TT), 128, 0, stream>>>(
        xsall, l * HH, srcs, dsts,
        preW + (size_t)l * TT * 32 * FF, preb + (size_t)l * TT * FF,
        s1, s2, mxk, mnk);
    k_agg<<<(NN * HH + 255) / 256, 256, 0, stream>>>(deg, s1, s2, mxk, mnk, aggb);
    k_post<<<NN / 16, 128, 0, stream>>>(
        xsall, l * HH, aggb, amp,
        postW + (size_t)l * TT * KPOST * FF, postb + (size_t)l * TT * FF, post);
    k_lin<<<NN / 16, 128, 0, stream>>>(
        post, linW + (size_t)l * HH * HH, linb + (size_t)l * HH,
        gamma + (size_t)l * HH, beta + (size_t)l * HH, xsall, (l + 1) * HH);
  }

  k_cnt<<<(NN + 255) / 256, 256, 0, stream>>>(batch, cnt);
  k_pool<<<(NN * HJ + 255) / 256, 256, 0, stream>>>(xsall, batch, gsum, gmaxk);
  k_gbuild<<<(GG * 3 * HJ + 255) / 256, 256, 0, stream>>>(gsum, gmaxk, cnt, gbuf);
  k_mlp1<<<(GG * HH + 255) / 256, 256, 0, stream>>>(gbuf, W1, b1, hidden);
  k_mlp2<<<1, 32, 0, stream>>>(hidden, W2, b2, out);
}